// TopK_7713761264047
// MI455X (gfx1250) — compile-verified
//
#include <hip/hip_runtime.h>
#include <hip/hip_bf16.h>

// Top-k(k=64) per row -> relu -> scatter into zeros.
// B=8192 rows, D=24576 cols, fp32. One workgroup per row.
// TDM (tensor_load_to_lds) stages the 96KB row into LDS once; 4x 8-bit MSB
// radix-select (sub-bin privatized LDS histogram + parallel suffix scan)
// finds the exact k-th key; deterministic index-ranked tie-break; single
// coalesced nontemporal b128 output pass. HBM traffic = 1R + 1W (~69us floor).

#define TK_D      24576
#define TK_D4     (TK_D / 4)
#define TK_BLOCK  512
#define TK_SUB    4          // histogram sub-bins (contention spreading)
#define TK_TIECAP 128        // deterministic tie-rank capacity

typedef __attribute__((ext_vector_type(4))) unsigned int uint32x4_t;
typedef __attribute__((ext_vector_type(8))) int          int32x8_t;
typedef __attribute__((ext_vector_type(4))) int          int32x4_t;
typedef __attribute__((ext_vector_type(4))) float        f32x4;   // native vec

#if defined(__gfx1250__) && __has_builtin(__builtin_amdgcn_tensor_load_to_lds) && \
    __has_builtin(__builtin_amdgcn_s_wait_tensorcnt)
#define TK_USE_TDM 1
#else
#define TK_USE_TDM 0
#endif

// Monotone float -> uint key: larger float => larger key.
__device__ __forceinline__ unsigned f2key(float f) {
    unsigned u = __float_as_uint(f);
    return (u & 0x80000000u) ? ~u : (u | 0x80000000u);
}

__device__ __forceinline__ float select_out(float v, unsigned idx, unsigned T,
                                            unsigned nties, const unsigned* tieidx,
                                            unsigned tcnt, unsigned* claim) {
    unsigned key = f2key(v);
    if (key > T) return fmaxf(v, 0.0f);
    if (key != T) return 0.0f;
    if (tcnt <= TK_TIECAP) {
        // Deterministic: rank ties by element index (order-independent of the
        // atomic append order that built the list).
        unsigned rank = 0;
        for (unsigned j = 0; j < tcnt; ++j) rank += (tieidx[j] < idx) ? 1u : 0u;
        return (rank < nties) ? fmaxf(v, 0.0f) : 0.0f;
    }
    // Pathological overflow fallback (practically unreachable for normal data).
    unsigned slot = atomicAdd(claim, 1u);
    return (slot < nties) ? fmaxf(v, 0.0f) : 0.0f;
}

__global__ void __launch_bounds__(TK_BLOCK)
topk_scatter_kernel(const float* __restrict__ x, const int* __restrict__ kptr,
                    float* __restrict__ out) {
    __shared__ float    row[TK_D];                 // 96 KB row cache
    __shared__ unsigned hist[256][TK_SUB];         // 4 KB privatized histogram
    __shared__ unsigned hsum[256];                 // 1 KB reduced bins / suffix scan
    __shared__ unsigned tieidx[TK_TIECAP];         // 512 B tie index list
    __shared__ unsigned ctrl[4];                   // [0]=prefix [1]=remaining
                                                   // [2]=tie count [3]=claim fallback

    const int tid = threadIdx.x;
    const size_t rbase = (size_t)blockIdx.x * (size_t)TK_D;
    const float* __restrict__ xr  = x + rbase;
    float* __restrict__ outr      = out + rbase;
    const int k = *kptr;

    // ---------------- Stage row into LDS ----------------
#if TK_USE_TDM
    if ((tid >> 5) == 0) {  // wave 0 only: TDM is a per-wave DMA op (EXEC ignored)
        unsigned lds_addr = (unsigned)(size_t)(&row[0]);
        unsigned long long ga = (unsigned long long)(const void*)xr;

        // D# group 0 (128b): count=1 | lds_addr | global_addr[56:0] | type=2
        uint32x4_t g0;
        g0.x = 1u;                                   // count=1, user mode
        g0.y = lds_addr;                             // lds_addr [63:32]
        g0.z = (unsigned)(ga & 0xFFFFFFFFu);         // global_addr[31:0]
        g0.w = (unsigned)((ga >> 32) & 0x01FFFFFFu)  // global_addr[56:32]
               | (2u << 30);                         // type = 2 ("image")

        // D# group 1 (256b): data_size=4B, 2D: dim0=D, dim1=1, tile=Dx1.
        int32x8_t g1;
        g1[0] = (int)(2u << 16);                                  // data_size=2 (4B)
        g1[1] = (int)(((unsigned)TK_D & 0xFFFFu) << 16);          // tensor_dim0[15:0]
        g1[2] = (int)((((unsigned)TK_D >> 16) & 0xFFFFu)          // tensor_dim0[31:16]
                      | (1u << 16));                              // tensor_dim1[15:0]=1
        g1[3] = (int)(((unsigned)TK_D & 0xFFFFu) << 16);          // tile_dim0=D
        g1[4] = (int)1;                                           // tile_dim1=1
        g1[5] = (int)TK_D;                                        // dim0_stride[31:0]
        g1[6] = 0;
        g1[7] = 0;

        int32x4_t gz4 = {0, 0, 0, 0};
        int32x8_t gz8 = {0, 0, 0, 0, 0, 0, 0, 0};
        __builtin_amdgcn_tensor_load_to_lds(g0, g1, gz4, gz4, gz8, 0);
        __builtin_amdgcn_s_wait_tensorcnt(0);
    }
#else
    {
        const f32x4* __restrict__ x4 = reinterpret_cast<const f32x4*>(xr);
        f32x4* __restrict__ r4 = reinterpret_cast<f32x4*>(row);
        for (int i = tid; i < TK_D4; i += TK_BLOCK)
            r4[i] = __builtin_nontemporal_load(&x4[i]);
    }
#endif
    if (tid == 0) { ctrl[2] = 0u; ctrl[3] = 0u; }
    __syncthreads();

    // ---------------- Radix select: find exact k-th largest key ----------------
    const f32x4* __restrict__ row4 = reinterpret_cast<const f32x4*>(row);
    unsigned prefix = 0u;
    unsigned remaining = (unsigned)k;

    for (int shift = 24; shift >= 0; shift -= 8) {
        unsigned* hflat = &hist[0][0];
        for (int i = tid; i < 256 * TK_SUB; i += TK_BLOCK) hflat[i] = 0u;
        __syncthreads();

        const unsigned mask = (shift == 24) ? 0u : (0xFFFFFFFFu << (shift + 8));
        const unsigned want = prefix & mask;
        const unsigned sub  = (unsigned)tid & (TK_SUB - 1);

        for (int i = tid; i < TK_D4; i += TK_BLOCK) {
            f32x4 v = row4[i];                        // ds_load_b128
            unsigned ks0 = f2key(v.x), ks1 = f2key(v.y);
            unsigned ks2 = f2key(v.z), ks3 = f2key(v.w);
            if ((ks0 & mask) == want) atomicAdd(&hist[(ks0 >> shift) & 255u][sub], 1u);
            if ((ks1 & mask) == want) atomicAdd(&hist[(ks1 >> shift) & 255u][sub], 1u);
            if ((ks2 & mask) == want) atomicAdd(&hist[(ks2 >> shift) & 255u][sub], 1u);
            if ((ks3 & mask) == want) atomicAdd(&hist[(ks3 >> shift) & 255u][sub], 1u);
        }
        __syncthreads();

        // Reduce sub-bins, then parallel suffix-sum over 256 bins (8 log-steps).
        if (tid < 256)
            hsum[tid] = hist[tid][0] + hist[tid][1] + hist[tid][2] + hist[tid][3];
        __syncthreads();
        for (int off = 1; off < 256; off <<= 1) {
            unsigned v = 0u;
            if (tid < 256 && tid + off < 256) v = hsum[tid + off];
            __syncthreads();
            if (tid < 256) hsum[tid] += v;
            __syncthreads();
        }
        // hsum[t] = count of matching keys with digit >= t. Unique t satisfies:
        if (tid < 256) {
            unsigned above = (tid == 255) ? 0u : hsum[tid + 1];
            if (hsum[tid] >= remaining && above < remaining) {
                ctrl[0] = prefix | ((unsigned)tid << shift);
                ctrl[1] = remaining - above;   // budget remaining inside bin tid
            }
        }
        __syncthreads();
        prefix = ctrl[0];
        remaining = ctrl[1];
    }

    const unsigned T = prefix;          // exact k-th largest key
    const unsigned nties = remaining;   // how many key==T elements to keep

    // ---------------- Collect tie indices (deterministic rank basis) ----------
    for (int i = tid; i < TK_D4; i += TK_BLOCK) {
        f32x4 v = row4[i];
        unsigned base = 4u * (unsigned)i;
        if (f2key(v.x) == T) { unsigned p = atomicAdd(&ctrl[2], 1u); if (p < TK_TIECAP) tieidx[p] = base + 0u; }
        if (f2key(v.y) == T) { unsigned p = atomicAdd(&ctrl[2], 1u); if (p < TK_TIECAP) tieidx[p] = base + 1u; }
        if (f2key(v.z) == T) { unsigned p = atomicAdd(&ctrl[2], 1u); if (p < TK_TIECAP) tieidx[p] = base + 2u; }
        if (f2key(v.w) == T) { unsigned p = atomicAdd(&ctrl[2], 1u); if (p < TK_TIECAP) tieidx[p] = base + 3u; }
    }
    __syncthreads();
    const unsigned tcnt = ctrl[2];

    // ---------------- Output pass: full row, nontemporal b128 stores ----------
    f32x4* __restrict__ out4 = reinterpret_cast<f32x4*>(outr);
    for (int i = tid; i < TK_D4; i += TK_BLOCK) {
        f32x4 v = row4[i];
        unsigned base = 4u * (unsigned)i;
        f32x4 o;
        o.x = select_out(v.x, base + 0u, T, nties, tieidx, tcnt, &ctrl[3]);
        o.y = select_out(v.y, base + 1u, T, nties, tieidx, tcnt, &ctrl[3]);
        o.z = select_out(v.z, base + 2u, T, nties, tieidx, tcnt, &ctrl[3]);
        o.w = select_out(v.w, base + 3u, T, nties, tieidx, tcnt, &ctrl[3]);
        __builtin_nontemporal_store(o, &out4[i]);     // global_store_b128 th:NT
    }
}

extern "C" void kernel_launch(void* const* d_in, const int* in_sizes, int n_in,
                              void* d_out, int out_size, void* d_ws, size_t ws_size,
                              hipStream_t stream) {
    const float* x  = (const float*)d_in[0];
    const int*   kp = (const int*)d_in[1];
    float* out = (float*)d_out;

    int rows = in_sizes[0] / TK_D;   // 8192
    dim3 grid((unsigned)rows);
    dim3 block(TK_BLOCK);
    topk_scatter_kernel<<<grid, block, 0, stream>>>(x, kp, out);
}